// Context_CP_22204980920540
// MI455X (gfx1250) — compile-verified
//
#include <hip/hip_runtime.h>

#define RANK   256
#define NB     50
#define NENT   100000
#define BATCH  1024

typedef __attribute__((ext_vector_type(2))) float v2f;
typedef __attribute__((ext_vector_type(8))) float v8f;

// async-load builtins take pointers to GCC-style int vectors (per hipcc diag)
typedef int gv4i __attribute__((vector_size(16)));
typedef int gv2i __attribute__((vector_size(8)));

#if __has_builtin(__builtin_amdgcn_global_load_async_to_lds_b128) && \
    __has_builtin(__builtin_amdgcn_global_load_async_to_lds_b64)
#define HAVE_ASYNC_LDS 1
#else
#define HAVE_ASYNC_LDS 0
#endif

#define AS1V4(p) ((__attribute__((address_space(1))) gv4i*)(p))
#define AS3V4(p) ((__attribute__((address_space(3))) gv4i*)(p))
#define AS1V2(p) ((__attribute__((address_space(1))) gv2i*)(p))
#define AS3V2(p) ((__attribute__((address_space(3))) gv2i*)(p))

__device__ __forceinline__ void wait_async_13() {
#if HAVE_ASYNC_LDS
#if __has_builtin(__builtin_amdgcn_s_wait_asynccnt)
  __builtin_amdgcn_s_wait_asynccnt(13);
#else
  asm volatile("s_wait_asynccnt 13" ::: "memory");
#endif
#endif
}
__device__ __forceinline__ void wait_async_0() {
#if HAVE_ASYNC_LDS
#if __has_builtin(__builtin_amdgcn_s_wait_asynccnt)
  __builtin_amdgcn_s_wait_asynccnt(0);
#else
  asm volatile("s_wait_asynccnt 0" ::: "memory");
#endif
#endif
}

// ---------------------------------------------------------------------------
// Kernel 1: per-row prep. One block (256 threads, 8 wave32) per batch row.
// ---------------------------------------------------------------------------
__global__ __launch_bounds__(256) void prep_kernel(
    const int*   __restrict__ x,       // [B,3]
    const int*   __restrict__ nb_idx,  // [B,NB]
    const float* __restrict__ lhs_w,   // [N_ENT,RANK]
    const float* __restrict__ rel_w,   // [N_REL,RANK]
    const float* __restrict__ rhs_w,   // [N_ENT,RANK]
    const float* __restrict__ W_w,     // [RANK,2*RANK]
    const float* __restrict__ W_b,     // [RANK]
    const float* __restrict__ W2_w,    // [RANK,RANK]
    const float* __restrict__ W2_b,    // [RANK]
    const float* __restrict__ Wo_w,    // [1,RANK]
    const float* __restrict__ Wo_b,    // [1]
    const float* __restrict__ Uo_w,    // [1,RANK]
    const float* __restrict__ Uo_b,    // [1]
    float* __restrict__ V,             // [B,RANK] workspace (GEMM A)
    float* __restrict__ out_tail)      // lhs|rel|rhs|e_c, each [B,RANK]
{
  __shared__ float trp[2 * RANK];   // [lhs | rel]
  __shared__ float wv[RANK];
  __shared__ float score[NB];
  __shared__ float alpha[NB];
  __shared__ float ecp[RANK];
  __shared__ float red[8];
  __shared__ float gsh;

  const int b    = blockIdx.x;
  const int k    = threadIdx.x;       // 0..255
  const int wave = threadIdx.x >> 5;  // wave32
  const int lane = threadIdx.x & 31;

  const int i0 = x[3 * b + 0];
  const int i1 = x[3 * b + 1];
  const int i2 = x[3 * b + 2];

  const float lhs = lhs_w[(size_t)i0 * RANK + k];
  const float rel = rel_w[(size_t)i1 * RANK + k];
  const float rhs = rhs_w[(size_t)i2 * RANK + k];
  trp[k]        = lhs;
  trp[RANK + k] = rel;
  __syncthreads();

  // w[k] = W_b[k] + sum_j trp[j] * W_w[k, j]
  {
    const float4* Wrow = (const float4*)(W_w + (size_t)k * (2 * RANK));
    float acc = W_b[k];
#pragma unroll 4
    for (int j4 = 0; j4 < (2 * RANK) / 4; ++j4) {
      float4 wq = Wrow[j4];
      acc += wq.x * trp[4 * j4 + 0] + wq.y * trp[4 * j4 + 1] +
             wq.z * trp[4 * j4 + 2] + wq.w * trp[4 * j4 + 3];
    }
    wv[k] = acc;
  }
  __syncthreads();

  // attention scores
  for (int m = wave; m < NB; m += 8) {
    const float* nrow = rhs_w + (size_t)nb_idx[b * NB + m] * RANK;
    float p = 0.f;
#pragma unroll
    for (int kk = 0; kk < RANK / 32; ++kk)
      p += wv[lane + 32 * kk] * nrow[lane + 32 * kk];
    for (int off = 16; off > 0; off >>= 1)
      p += __shfl_xor(p, off, 32);
    if (lane == 0) score[m] = p;
  }
  __syncthreads();

  if (threadIdx.x == 0) {
    float mx = score[0];
    for (int m = 1; m < NB; ++m) mx = fmaxf(mx, score[m]);
    float s = 0.f;
    for (int m = 0; m < NB; ++m) { float e = __expf(score[m] - mx); alpha[m] = e; s += e; }
    float inv = 1.f / s;
    for (int m = 0; m < NB; ++m) alpha[m] *= inv;
  }
  __syncthreads();

  // e_c_pre[k]
  float ep = 0.f;
  for (int m = 0; m < NB; ++m)
    ep += alpha[m] * rhs_w[(size_t)nb_idx[b * NB + m] * RANK + k];
  ecp[k] = ep;
  __syncthreads();

  // e_c[k] = W2_b[k] + sum_j e_c_pre[j] * W2_w[k, j]
  float ec;
  {
    const float4* Wrow = (const float4*)(W2_w + (size_t)k * RANK);
    float acc = W2_b[k];
#pragma unroll 4
    for (int j4 = 0; j4 < RANK / 4; ++j4) {
      float4 wq = Wrow[j4];
      acc += wq.x * ecp[4 * j4 + 0] + wq.y * ecp[4 * j4 + 1] +
             wq.z * ecp[4 * j4 + 2] + wq.w * ecp[4 * j4 + 3];
    }
    ec = acc;
  }

  // gate
  float part = lhs * rel * Uo_w[k] + ec * Wo_w[k];
  for (int off = 16; off > 0; off >>= 1)
    part += __shfl_xor(part, off, 32);
  if (lane == 0) red[wave] = part;
  __syncthreads();
  if (threadIdx.x == 0) {
    float s = Uo_b[0] + Wo_b[0];
    for (int w8 = 0; w8 < 8; ++w8) s += red[w8];
    gsh = 1.f / (1.f + __expf(-s));
  }
  __syncthreads();
  const float g = gsh;

  const float v = lhs * rel * (g * ec + (1.f - g));
  V[(size_t)b * RANK + k] = v;

  out_tail[(size_t)b * RANK + k]                            = lhs;
  out_tail[(size_t)BATCH * RANK     + (size_t)b * RANK + k] = rel;
  out_tail[(size_t)2 * BATCH * RANK + (size_t)b * RANK + k] = rhs;
  out_tail[(size_t)3 * BATCH * RANK + (size_t)b * RANK + k] = ec;
}

// ---------------------------------------------------------------------------
// Kernel 2: out[b,n] = sum_k V[b,k] * rhs_w[n,k]   (1024 x 100000, K=256)
// fp32 WMMA 16x16x4. Block tile M=256 x N=80, K in 32-chunks, double-buffered
// LDS filled by GLOBAL_LOAD_ASYNC_TO_LDS (ASYNCcnt), prefetching chunk c+1
// while computing chunk c. 8 wave32 per block; wave w owns M rows
// [32w, 32w+32) as two 16-row A frags x five 16-col B frags = 10 WMMA/K-step.
// ---------------------------------------------------------------------------
#define MT 256
#define NT 80
#define KT 32
#define KST (KT + 4)          // 36-float row stride: conflict-free frag reads
#define NCHUNK (RANK / KT)    // 8
// per-thread async issues per chunk: As 2048 b128 /256 = 8, Bs 1280 b64 /256 = 5

__global__ __launch_bounds__(256) void gemm_kernel(
    const float* __restrict__ V,      // [BATCH, RANK]
    const float* __restrict__ rhs_w,  // [NENT, RANK]
    float* __restrict__ out)          // [BATCH, NENT]
{
  __shared__ float As[2][MT][KST];
  __shared__ float Bs[2][NT][KST];

  const int n0   = blockIdx.x * NT;
  const int m0   = blockIdx.y * MT;
  const int tid  = threadIdx.x;
  const int wave = tid >> 5;
  const int lane = tid & 31;
  const int l15  = lane & 15;
  const int kb0  = (lane >> 4) << 1;   // K sub-offset 0 / 2 per ISA A/B layout

  v8f acc[2][5] = {};

  auto issue_chunk = [&](int c, int buf) {
    const int k0 = c * KT;
#if HAVE_ASYNC_LDS
#pragma unroll
    for (int j = 0; j < 8; ++j) {             // As: 256x32 floats as b128
      int i = tid + 256 * j;
      int r = i >> 3, c4 = (i & 7) << 2;
      __builtin_amdgcn_global_load_async_to_lds_b128(
          AS1V4(V + (size_t)(m0 + r) * RANK + k0 + c4),
          AS3V4(&As[buf][r][c4]), 0, 0);
    }
#pragma unroll
    for (int j = 0; j < 5; ++j) {             // Bs: 80x32 floats as b64
      int i = tid + 256 * j;
      int r = i >> 4, c2 = (i & 15) << 1;
      __builtin_amdgcn_global_load_async_to_lds_b64(
          AS1V2(rhs_w + (size_t)(n0 + r) * RANK + k0 + c2),
          AS3V2(&Bs[buf][r][c2]), 0, 0);
    }
#else
#pragma unroll
    for (int j = 0; j < 8; ++j) {
      int i = tid + 256 * j;
      int r = i >> 3, c4 = (i & 7) << 2;
      *(float4*)&As[buf][r][c4] =
          *(const float4*)(V + (size_t)(m0 + r) * RANK + k0 + c4);
    }
#pragma unroll
    for (int j = 0; j < 5; ++j) {
      int i = tid + 256 * j;
      int r = i >> 4, c2 = (i & 15) << 1;
      *(float2*)&Bs[buf][r][c2] =
          *(const float2*)(rhs_w + (size_t)(n0 + r) * RANK + k0 + c2);
    }
#endif
  };

  auto compute_chunk = [&](int buf) {
#pragma unroll
    for (int kk = 0; kk < KT; kk += 4) {
      v2f a0 = *(const v2f*)&As[buf][(wave << 5) + l15][kk + kb0];
      v2f a1 = *(const v2f*)&As[buf][(wave << 5) + 16 + l15][kk + kb0];
      v2f bf[5];
#pragma unroll
      for (int t = 0; t < 5; ++t)
        bf[t] = *(const v2f*)&Bs[buf][(t << 4) + l15][kk + kb0];
#pragma unroll
      for (int t = 0; t < 5; ++t) {
        acc[0][t] = __builtin_amdgcn_wmma_f32_16x16x4_f32(
            false, a0, false, bf[t], (short)0, acc[0][t], false, false);
        acc[1][t] = __builtin_amdgcn_wmma_f32_16x16x4_f32(
            false, a1, false, bf[t], (short)0, acc[1][t], false, false);
      }
    }
  };

#if HAVE_ASYNC_LDS
  issue_chunk(0, 0);
  for (int c = 0; c < NCHUNK; ++c) {
    const int cur = c & 1;
    if (c + 1 < NCHUNK) {
      issue_chunk(c + 1, cur ^ 1);  // prefetch next chunk into other buffer
      wait_async_13();              // in-order: drains the 13 ops of chunk c
    } else {
      wait_async_0();
    }
    __syncthreads();                // all waves' chunk-c data resident
    compute_chunk(cur);
    __syncthreads();                // safe to overwrite buf cur next round
  }
#else
  for (int c = 0; c < NCHUNK; ++c) {
    const int cur = c & 1;
    issue_chunk(c, cur);
    __syncthreads();
    compute_chunk(cur);
    __syncthreads();
  }
#endif

  // C/D layout: VGPR r -> row r + 8*(lane>=16), col = lane&15
  const int rowbase = m0 + (wave << 5) + ((lane >> 4) << 3);
#pragma unroll
  for (int mi = 0; mi < 2; ++mi) {
#pragma unroll
    for (int t = 0; t < 5; ++t) {
#pragma unroll
      for (int r = 0; r < 8; ++r) {
        int row = rowbase + (mi << 4) + r;
        __builtin_nontemporal_store(
            acc[mi][t][r], out + (size_t)row * NENT + n0 + (t << 4) + l15);
      }
    }
  }
}

// ---------------------------------------------------------------------------
extern "C" void kernel_launch(void* const* d_in, const int* in_sizes, int n_in,
                              void* d_out, int out_size, void* d_ws, size_t ws_size,
                              hipStream_t stream) {
  const int*   x      = (const int*)d_in[0];
  const int*   nb_idx = (const int*)d_in[1];
  const float* lhs_w  = (const float*)d_in[2];
  const float* rel_w  = (const float*)d_in[3];
  const float* rhs_w  = (const float*)d_in[4];
  const float* W_w    = (const float*)d_in[5];
  const float* W_b    = (const float*)d_in[6];
  const float* W2_w   = (const float*)d_in[7];
  const float* W2_b   = (const float*)d_in[8];
  const float* Wo_w   = (const float*)d_in[9];
  const float* Wo_b   = (const float*)d_in[10];
  const float* Uo_w   = (const float*)d_in[11];
  const float* Uo_b   = (const float*)d_in[12];

  float* out      = (float*)d_out;
  float* V        = (float*)d_ws;                  // 1 MB scratch
  float* out_tail = out + (size_t)BATCH * NENT;    // lhs|rel|rhs|e_c

  prep_kernel<<<BATCH, 256, 0, stream>>>(
      x, nb_idx, lhs_w, rel_w, rhs_w, W_w, W_b, W2_w, W2_b,
      Wo_w, Wo_b, Uo_w, Uo_b, V, out_tail);

  dim3 g2(NENT / NT, BATCH / MT);  // 1250 x 4
  gemm_kernel<<<g2, 256, 0, stream>>>(V, rhs_w, out);
}